// QDense4StatesUndirected_78142634983552
// MI455X (gfx1250) — compile-verified
//
#include <hip/hip_runtime.h>
#include <math.h>

// Quantum circuit simulator: 10 wires (1024 amplitudes), QDEPTH=8 layers,
// BATCH=2048. State resident in LDS per workgroup. All single-qubit gates are
// fused into 16x16 complex matrices applied with V_WMMA_F32_16X16X4_F32
// (three stages per layer: bits 3..0, bits 7..4, bits {9,8}+{1,0} with the
// 4x4 gate embedded as GC4 (x) I4). The CNOT ladder is composed into a single
// LDS gather permutation per layer. Initial x tile is brought in with
// global_load_async_to_lds_b128 (ASYNCcnt path).

#define NWIRES   10
#define NSTATE   1024
#define NQDEPTH  8
#define ROWS     4          // batch rows per workgroup
#define NTHREADS 256        // 8 waves (wave32)
#define NBATCH   2048

typedef __attribute__((ext_vector_type(2))) float v2f;
typedef __attribute__((ext_vector_type(8))) float v8f;

// element index within a 1024-amplitude state for (row m, group value k):
//  MODE 0: gate on bits 3..0   -> e = m*16 + k           (m = bits 9..4)
//  MODE 1: gate on bits 7..4   -> e = k*16 + m           (m = bits 3..0 | 9..8 via base)
//  MODE 2: gate on bits 9,8,1,0-> e = (k>>2)*256 + m*4 + (k&3)   (m = bits 7..2)
template <int MODE>
__device__ __forceinline__ int elem_of(int m, int k) {
  if (MODE == 0) return m * 16 + k;
  if (MODE == 1) return k * 16 + m;
  return ((k >> 2) << 8) + (m << 2) + (k & 3);
}

// Apply a 16x16 complex gate (Gr + i*Gi, row-major [i_new][j_old]) to the
// LDS-resident state planes. 16 tiles per workgroup, 2 per wave.
template <int MODE>
__device__ __forceinline__ void apply16(float* sRe, float* sIm,
                                        const float* Gr, const float* Gi,
                                        int lane, int wv)
{
  const int n  = lane & 15;          // B-fragment column / D column
  const int m  = lane & 15;          // A-fragment row
  const int kp = (lane >> 4) << 1;   // K sub-pair select (0 or 2)
  const int blksz = (MODE == 2) ? 64 : 256;

#if __has_builtin(__builtin_amdgcn_wmma_f32_16x16x4_f32)
  // B fragments (gate), constant across tiles: B[k][n] = G[n][k]
  v2f Br[4], Bi2[4], Bni[4];
#pragma unroll
  for (int k4 = 0; k4 < 4; ++k4) {
    int k0 = k4 * 4 + kp;
    Br [k4].x = Gr[n * 16 + k0];  Br [k4].y = Gr[n * 16 + k0 + 1];
    Bi2[k4].x = Gi[n * 16 + k0];  Bi2[k4].y = Gi[n * 16 + k0 + 1];
    Bni[k4].x = -Bi2[k4].x;       Bni[k4].y = -Bi2[k4].y;   // float WMMA has no A/B NEG
  }
#endif

  const int nt = (ROWS * 4) / 8;   // tiles per wave
#pragma unroll
  for (int j = 0; j < nt; ++j) {
    const int t    = wv * nt + j;
    const int base = (t >> 2) * NSTATE + (t & 3) * blksz;

    v8f Dr = {0.f,0.f,0.f,0.f,0.f,0.f,0.f,0.f};
    v8f Di = {0.f,0.f,0.f,0.f,0.f,0.f,0.f,0.f};

#if __has_builtin(__builtin_amdgcn_wmma_f32_16x16x4_f32)
    // A fragments: the full 256-element tile is read before any store.
    v2f Ar[4], Ai[4];
#pragma unroll
    for (int k4 = 0; k4 < 4; ++k4) {
      int k0 = k4 * 4 + kp;
      int e0 = elem_of<MODE>(m, k0);
      int e1 = elem_of<MODE>(m, k0 + 1);
      Ar[k4].x = sRe[base + e0]; Ar[k4].y = sRe[base + e1];
      Ai[k4].x = sIm[base + e0]; Ai[k4].y = sIm[base + e1];
    }
    // (Xr + iXi) * (Gr^T + iGi^T): 4 real GEMMs, 16 WMMAs.
#pragma unroll
    for (int k4 = 0; k4 < 4; ++k4) {
      Dr = __builtin_amdgcn_wmma_f32_16x16x4_f32(false, Ar[k4], false, Br [k4], (short)0, Dr, false, false);
      Dr = __builtin_amdgcn_wmma_f32_16x16x4_f32(false, Ai[k4], false, Bni[k4], (short)0, Dr, false, false);
      Di = __builtin_amdgcn_wmma_f32_16x16x4_f32(false, Ar[k4], false, Bi2[k4], (short)0, Di, false, false);
      Di = __builtin_amdgcn_wmma_f32_16x16x4_f32(false, Ai[k4], false, Br [k4], (short)0, Di, false, false);
    }
#else
    // VALU fallback
#pragma unroll
    for (int v = 0; v < 8; ++v) {
      int M = v + ((lane >> 4) << 3), N = lane & 15;
      float accr = 0.f, acci = 0.f;
#pragma unroll
      for (int k = 0; k < 16; ++k) {
        int e = elem_of<MODE>(M, k);
        float ar = sRe[base + e], ai = sIm[base + e];
        float gr = Gr[N * 16 + k], gi = Gi[N * 16 + k];
        accr += ar * gr - ai * gi;
        acci += ar * gi + ai * gr;
      }
      Dr[v] = accr; Di[v] = acci;
    }
#endif
    // Write back: D row M = vgpr + 8*(lane>=16), col N = lane&15.
#pragma unroll
    for (int v = 0; v < 8; ++v) {
      int M = v + ((lane >> 4) << 3);
      int N = lane & 15;
      int e = elem_of<MODE>(M, N);
      sRe[base + e] = Dr[v];
      sIm[base + e] = Di[v];
    }
  }
}

__global__ __launch_bounds__(NTHREADS)
void qcircuit_kernel(const float* __restrict__ x, const float* __restrict__ w,
                     const int* __restrict__ repsPtr, float* __restrict__ out)
{
  __shared__ float sRe[ROWS * NSTATE];
  __shared__ float sIm[ROWS * NSTATE];
  __shared__ float gAr[256], gAi[256];   // wires 6..9 (state bits 3..0)
  __shared__ float gBr[256], gBi[256];   // wires 2..5 (state bits 7..4)
  __shared__ float gCr[256], gCi[256];   // wires 0..1 on bits 9,8 (x) I4 on bits 1,0
  __shared__ float uRe[NWIRES * 4], uIm[NWIRES * 4];
  __shared__ float red[NTHREADS];
  __shared__ float norms[ROWS];

  const int tid  = threadIdx.x;
  const int wg   = blockIdx.x;
  const int lane = tid & 31;
  const int wv   = tid >> 5;

  // ---- async-copy x tile into the real plane (ASYNCcnt path) ----
  const float* xg = x + (size_t)wg * ROWS * NSTATE;
  {
    unsigned ldsRe = (unsigned)(size_t)(&sRe[0]);    // LDS byte offset (low 32 bits)
#pragma unroll
    for (int k = 0; k < (ROWS * NSTATE) / (NTHREADS * 4); ++k) {
      unsigned ebytes = (unsigned)(tid * 4 + k * NTHREADS * 4) * 4u;  // 16B per lane
      asm volatile("global_load_async_to_lds_b128 %0, %1, %2"
                   :: "v"(ldsRe + ebytes), "v"(ebytes), "s"(xg) : "memory");
    }
    for (int idx = tid; idx < ROWS * NSTATE; idx += NTHREADS) sIm[idx] = 0.f;
    asm volatile("s_wait_asynccnt 0x0" ::: "memory");
  }
  __syncthreads();

  // ---- L2 norm per row, normalize ----
  {
    int r = tid >> 6, off = tid & 63;          // 64 threads per row
    float acc = 0.f;
    const float* rp = &sRe[r * NSTATE];
#pragma unroll 4
    for (int k = off; k < NSTATE; k += 64) { float v = rp[k]; acc += v * v; }
    red[tid] = acc;
  }
  __syncthreads();
  if (tid < ROWS) {
    float s = 0.f;
    for (int j = 0; j < 64; ++j) s += red[tid * 64 + j];
    float nm = sqrtf(s);
    norms[tid] = (nm > 0.f) ? nm : 1.f;
  }
  __syncthreads();
  for (int idx = tid; idx < ROWS * NSTATE; idx += NTHREADS)
    sRe[idx] *= 1.f / norms[idx >> 10];
  __syncthreads();

  const int reps = repsPtr[0];
  for (int rep = 0; rep < reps; ++rep) {
    for (int l = 0; l < NQDEPTH; ++l) {
      // ---- per-wire 2x2 Rot matrices: RZ(phi) RY(theta) RZ(omega) ----
      if (tid < NWIRES) {
        float phi   = tanhf(w[(l * NWIRES + tid) * 3 + 0]);
        float theta = tanhf(w[(l * NWIRES + tid) * 3 + 1]);
        float omega = tanhf(w[(l * NWIRES + tid) * 3 + 2]);
        float c  = cosf(0.5f * theta), sn = sinf(0.5f * theta);
        float ap = 0.5f * (phi + omega), am = 0.5f * (phi - omega);
        float epr = cosf(ap), epi = -sinf(ap);   // e^{-i(phi+omega)/2}
        float emr = cosf(am), emi = -sinf(am);   // e^{-i(phi-omega)/2}
        int b = tid * 4;
        uRe[b + 0] =  epr * c;  uIm[b + 0] =  epi * c;   // U00 = ep*c
        uRe[b + 1] = -emr * sn; uIm[b + 1] =  emi * sn;  // U01 = -conj(em)*sn
        uRe[b + 2] =  emr * sn; uIm[b + 2] =  emi * sn;  // U10 = em*sn
        uRe[b + 3] =  epr * c;  uIm[b + 3] = -epi * c;   // U11 = conj(ep)*c
      }
      __syncthreads();
      // ---- Kronecker-build the three 16x16 group gates (wire q <-> bit 9-q) ----
      {
        int i = tid >> 4, j = tid & 15;
        float ar = 1.f, ai = 0.f, br = 1.f, bi = 0.f;
#pragma unroll
        for (int gb = 0; gb < 4; ++gb) {
          int ib = (i >> gb) & 1, jb = (j >> gb) & 1;
          int ea = (9 - gb) * 4 + ib * 2 + jb;   // stage A: wires 9..6 on bits 0..3
          int eb = (5 - gb) * 4 + ib * 2 + jb;   // stage B: wires 5..2 on bits 4..7
          float xr = uRe[ea], xi = uIm[ea];
          float t  = ar * xr - ai * xi; ai = ar * xi + ai * xr; ar = t;
          xr = uRe[eb]; xi = uIm[eb];
          t  = br * xr - bi * xi; bi = br * xi + bi * xr; br = t;
        }
        gAr[tid] = ar; gAi[tid] = ai;
        gBr[tid] = br; gBi[tid] = bi;
        // stage C: (U0 on bit9) (x) (U1 on bit8) on group bits 3..2, identity on bits 1..0
        float cvr = 0.f, cvi = 0.f;
        if ((i & 3) == (j & 3)) {
          int ih = i >> 2, jh = j >> 2;
          int e0 = 0 * 4 + ((ih >> 1) & 1) * 2 + ((jh >> 1) & 1);
          int e1 = 1 * 4 + (ih & 1) * 2 + (jh & 1);
          float r0 = uRe[e0], m0 = uIm[e0], r1 = uRe[e1], m1 = uIm[e1];
          cvr = r0 * r1 - m0 * m1;
          cvi = r0 * m1 + m0 * r1;
        }
        gCr[tid] = cvr; gCi[tid] = cvi;
      }
      __syncthreads();

      // ---- three WMMA stages ----
      apply16<0>(sRe, sIm, gAr, gAi, lane, wv);
      __syncthreads();
      apply16<1>(sRe, sIm, gBr, gBi, lane, wv);
      __syncthreads();
      apply16<2>(sRe, sIm, gCr, gCi, lane, wv);
      __syncthreads();

      // ---- CNOT ladder composed into one gather permutation ----
      // out[i] = in[sigma_1(sigma_2(...sigma_10(i)))], sigma_q = CNOT(q, (q+r)%10)
      {
        const int r = (l % (NWIRES - 1)) + 1;
        unsigned srcs[4];
#pragma unroll
        for (int kk = 0; kk < 4; ++kk) {
          unsigned j = (unsigned)(tid + kk * 256) & 1023u;
          for (int q = NWIRES - 1; q >= 0; --q) {
            int pc = 9 - q;
            int pt = 9 - ((q + r) % NWIRES);
            j ^= ((j >> pc) & 1u) << pt;
          }
          srcs[kk] = j;
        }
        float tr[16], ti[16];
#pragma unroll
        for (int k = 0; k < 16; ++k) {
          int base = (k >> 2) * NSTATE;
          unsigned src = srcs[k & 3];
          tr[k] = sRe[base + src];
          ti[k] = sIm[base + src];
        }
        __syncthreads();
#pragma unroll
        for (int k = 0; k < 16; ++k) {
          int dst = (k >> 2) * NSTATE + (tid + (k & 3) * 256);
          sRe[dst] = tr[k];
          sIm[dst] = ti[k];
        }
        __syncthreads();
      }
    }
  }

  // ---- scale back by the original norm, write interleaved complex64 ----
  float* og = out + (size_t)wg * ROWS * NSTATE * 2;
  for (int idx = tid; idx < ROWS * NSTATE; idx += NTHREADS) {
    float nm = norms[idx >> 10];
    og[2 * idx + 0] = sRe[idx] * nm;
    og[2 * idx + 1] = sIm[idx] * nm;
  }
}

extern "C" void kernel_launch(void* const* d_in, const int* in_sizes, int n_in,
                              void* d_out, int out_size, void* d_ws, size_t ws_size,
                              hipStream_t stream) {
  (void)in_sizes; (void)n_in; (void)d_ws; (void)ws_size; (void)out_size;
  const float* x    = (const float*)d_in[0];   // (2048,1,32,32) f32
  const float* wgt  = (const float*)d_in[1];   // (8,10,3) f32
  const int*   reps = (const int*)d_in[2];     // scalar int
  float*       out  = (float*)d_out;           // complex64 -> interleaved f32 pairs

  dim3 grid(NBATCH / ROWS), block(NTHREADS);
  qcircuit_kernel<<<grid, block, 0, stream>>>(x, wgt, reps, out);
}